// CkConv1D_25821343383756
// MI455X (gfx1250) — compile-verified
//
#include <hip/hip_runtime.h>

typedef float v2f __attribute__((ext_vector_type(2)));
typedef float v8f __attribute__((ext_vector_type(8)));

#define T_LEN 1024
#define C_IN  4
#define H_DIM 32
#define O_OUT 2
#define FDIM  64   // 2*H features: [cos u | sin u]

// workspace layout (float offsets)
#define WS_TAB 0                       // cp[256] then sp[256]
#define WS_F   512                     // F [T][64]
#define WS_G0  (WS_F  + T_LEN*FDIM)    // G_o=0 [T][64]
#define WS_G1  (WS_G0 + T_LEN*FDIM)    // G_o=1 [T][64]
#define WS_XS  (WS_G1 + T_LEN*FDIM)    // Xsum [T]

// ---------------------------------------------------------------------------
// Kernel A: per-(o,c,h) phase table, with w2[h] folded in.
// phi = o*w1[h,2] + c*w1[h,1] + b1[h];  cp = w2[h]*cos(phi), sp = w2[h]*sin(phi)
// ---------------------------------------------------------------------------
__global__ void ck_prep_tab(const float* __restrict__ w1, const float* __restrict__ b1,
                            const float* __restrict__ w2, float* __restrict__ ws) {
    int tid = threadIdx.x;              // 256 threads = O*C*H
    int h = tid & 31;
    int c = (tid >> 5) & 3;
    int o = tid >> 7;
    float phi = (float)o * w1[h*3+2] + (float)c * w1[h*3+1] + b1[h];
    float s, cs;
    sincosf(phi, &s, &cs);
    float w = w2[h];
    ws[WS_TAB + tid]       = w * cs;    // cp
    ws[WS_TAB + 256 + tid] = w * s;     // sp
}

// ---------------------------------------------------------------------------
// Kernel B: build F[j], G0[j], G1[j], Xsum[j] for each position j.
// ---------------------------------------------------------------------------
__global__ void ck_prep_feat(const float* __restrict__ x, const float* __restrict__ t,
                             const float* __restrict__ w1, float* __restrict__ ws) {
    int j = blockIdx.x * blockDim.x + threadIdx.x;
    if (j >= T_LEN) return;
    float xj[C_IN];
    float xs = 0.f;
    #pragma unroll
    for (int c = 0; c < C_IN; ++c) { xj[c] = x[j*C_IN + c]; xs += xj[c]; }
    ws[WS_XS + j] = xs;

    float tj = t[j];
    float* F  = ws + WS_F  + j*FDIM;
    float* G0 = ws + WS_G0 + j*FDIM;
    float* G1 = ws + WS_G1 + j*FDIM;
    const float* cp = ws + WS_TAB;
    const float* sp = ws + WS_TAB + 256;

    for (int h = 0; h < H_DIM; ++h) {
        float u = tj * w1[h*3+0];
        float su, cu;
        sincosf(u, &su, &cu);
        F[h]      = cu;
        F[32 + h] = su;
        // o = 0
        {
            float A = 0.f, B = 0.f;
            #pragma unroll
            for (int c = 0; c < C_IN; ++c) {
                A += xj[c] * cp[c*32 + h];
                B += xj[c] * sp[c*32 + h];
            }
            G0[h]      = A*su + B*cu;   // P
            G0[32 + h] = B*su - A*cu;   // Q
        }
        // o = 1
        {
            float A = 0.f, B = 0.f;
            #pragma unroll
            for (int c = 0; c < C_IN; ++c) {
                A += xj[c] * cp[128 + c*32 + h];
                B += xj[c] * sp[128 + c*32 + h];
            }
            G1[h]      = A*su + B*cu;
            G1[32 + h] = B*su - A*cu;
        }
    }
}

// ---------------------------------------------------------------------------
// Kernel C: y[i,o] = F[i,:] . (sum_{j < tile} G_o[j,:])        (inter, VALU)
//                  + rowsum_{j'<=i'} (F_tile @ G_tile^T)[i',j'] (intra, WMMA f32)
//                  + b2 * prefix(Xsum)(i)
// grid = 64 row-tiles, block = 64 threads = 2 waves (wave per output channel).
// ---------------------------------------------------------------------------
__global__ void ck_causal_out(const float* __restrict__ ws_c, const float* __restrict__ b2,
                              float* __restrict__ out) {
    const int I    = blockIdx.x;          // 16-row tile index
    const int o    = threadIdx.x >> 5;    // wave id == output channel
    const int lane = threadIdx.x & 31;

    const float* F  = ws_c + WS_F;
    const float* G  = ws_c + (o == 0 ? WS_G0 : WS_G1);
    const float* Xs = ws_c + WS_XS;

    __shared__ float s_S[O_OUT][FDIM];
    __shared__ float s_intra[O_OUT][16];

    // --- 1) inter-tile running feature sum over all rows before this tile ---
    float s0 = 0.f, s1 = 0.f;
    const int jmax = I * 16;
    for (int j = 0; j < jmax; ++j) {
        s0 += G[j*FDIM + lane];           // coalesced: lanes cover 64 features
        s1 += G[j*FDIM + lane + 32];
    }
    s_S[o][lane]      = s0;
    s_S[o][lane + 32] = s1;

    // --- 2) intra-tile 16x16 = F_tile(16x64) @ G_tile^T(64x16) via WMMA ----
    // A layout: lane(0-15)=M rows, K pair {0,1}; lanes(16-31)=K pair {2,3}.
    // B layout mirrors with lane%16 = N (row of G).
    const int row  = lane & 15;
    const int half = lane >> 4;
    const float* Fr = F + (I*16 + row) * FDIM + half*2;
    const float* Gr = G + (I*16 + row) * FDIM + half*2;

    v8f acc = {0.f,0.f,0.f,0.f,0.f,0.f,0.f,0.f};
    #pragma unroll
    for (int k0 = 0; k0 < FDIM; k0 += 4) {
        v2f a, b;
        a.x = Fr[k0]; a.y = Fr[k0+1];
        b.x = Gr[k0]; b.y = Gr[k0+1];
        acc = __builtin_amdgcn_wmma_f32_16x16x4_f32(
            /*neg_a=*/false, a, /*neg_b=*/false, b,
            /*c_mod=*/(short)0, acc, /*reuse_a=*/false, /*reuse_b=*/false);
    }

    // C/D layout: VGPR r -> M=r (lanes 0-15), M=8+r (lanes 16-31); N = lane%16.
    // Causal mask inside diagonal tile: keep N <= M, then row-sum across N.
    #pragma unroll
    for (int r = 0; r < 8; ++r) {
        int M = r + half*8;
        float v = (row <= M) ? acc[r] : 0.f;   // row == N here
        v += __shfl_xor(v, 1, 32);
        v += __shfl_xor(v, 2, 32);
        v += __shfl_xor(v, 4, 32);
        v += __shfl_xor(v, 8, 32);             // xor<16 stays within each half
        if (row == 0) s_intra[o][M] = v;
    }
    __syncthreads();

    // --- 3) per-row final output -------------------------------------------
    if (lane < 16) {
        const int i = I*16 + lane;
        float inter = 0.f;
        const float* Fi = F + i*FDIM;
        #pragma unroll 8
        for (int f = 0; f < FDIM; ++f) inter += Fi[f] * s_S[o][f];
        float p = 0.f;
        for (int j = 0; j <= i; ++j) p += Xs[j];
        out[i*O_OUT + o] = inter + s_intra[o][lane] + b2[0] * p;
    }
}

// ---------------------------------------------------------------------------
extern "C" void kernel_launch(void* const* d_in, const int* in_sizes, int n_in,
                              void* d_out, int out_size, void* d_ws, size_t ws_size,
                              hipStream_t stream) {
    const float* x  = (const float*)d_in[0];
    const float* t  = (const float*)d_in[1];
    const float* w1 = (const float*)d_in[2];
    const float* b1 = (const float*)d_in[3];
    const float* w2 = (const float*)d_in[4];
    const float* b2 = (const float*)d_in[5];
    float* ws  = (float*)d_ws;
    float* out = (float*)d_out;

    ck_prep_tab <<<1, 256, 0, stream>>>(w1, b1, w2, ws);
    ck_prep_feat<<<T_LEN/256, 256, 0, stream>>>(x, t, w1, ws);
    ck_causal_out<<<T_LEN/16, 64, 0, stream>>>(ws, b2, out);
}